// MixerModel_42228118455015
// MI455X (gfx1250) — compile-verified
//
#include <hip/hip_runtime.h>
#include <hip/hip_bf16.h>
#include <math.h>

// ---------------- problem constants ----------------
constexpr int Bsz  = 4;
constexpr int Lseq = 2048;
constexpr int DMd  = 1024;
constexpr int NL   = 4;
constexpr int DI   = 2 * DMd;      // 2048
constexpr int DS   = 16;
constexpr int DR   = DMd / 16;     // 64
constexpr int DC   = 4;
constexpr int Mrows = Bsz * Lseq;  // 8192
constexpr float EPS = 1e-5f;

typedef __attribute__((ext_vector_type(2))) float v2f;
typedef __attribute__((ext_vector_type(8))) float v8f;
typedef __attribute__((ext_vector_type(4))) int   v4i;

#if __has_builtin(__builtin_amdgcn_global_load_async_to_lds_b128)
#define USE_ASYNC_LDS 1
typedef __attribute__((address_space(1))) v4i* gbl_v4i_ptr;
typedef __attribute__((address_space(3))) v4i* lds_v4i_ptr;
#else
#define USE_ASYNC_LDS 0
#endif

// ---------------- generic fp32 WMMA GEMM, 32x32 per wave ----------------
// C[m][n] = sum_k A[m*lda + k] * Bw[n*ldb + k]   (C = A * Bw^T)
// Each wave32 owns a 32x32 output block = 2x2 WMMA tiles; per K-step (4)
// it loads 2 A-frags + 2 B-frags and issues 4 V_WMMA_F32_16X16X4_F32,
// reusing each fragment twice (halves L2 fetch traffic vs 1 tile/wave).
// A-frag layout (ISA 7.12.2): lane l<16 -> A[M=l][k0+0..1], lane l+16 ->
// A[M=l][k0+2..3]; B-frag mirrored. C/D: VGPR r: lanes0-15 M=r, 16-31 M=8+r.
__global__ void gemm_wmma_f32(const float* __restrict__ A,
                              const float* __restrict__ Bw,
                              float* __restrict__ C,
                              int M, int N, int K,
                              int lda, int ldb, int ldc, int accumulate) {
    const int lane = threadIdx.x;              // 0..31
    const int wv   = threadIdx.y;              // 0..3
    const int nt   = blockIdx.x * 4 + wv;      // 32-wide N tile
    const int mt   = blockIdx.y;               // 32-wide M tile
    if (nt * 32 >= N) return;

    const int l  = lane & 15;
    const int hi = lane >> 4;                  // selects K sub-pair

    const float* a0 = A  + (size_t)(mt * 32 + l) * lda + 2 * hi;
    const float* a1 = a0 + (size_t)16 * lda;
    const float* b0 = Bw + (size_t)(nt * 32 + l) * ldb + 2 * hi;
    const float* b1 = b0 + (size_t)16 * ldb;

    v8f acc00 = {}, acc01 = {}, acc10 = {}, acc11 = {};
    for (int k0 = 0; k0 < K; k0 += 4) {
        v2f fa0 = *(const v2f*)(a0 + k0);
        v2f fa1 = *(const v2f*)(a1 + k0);
        v2f fb0 = *(const v2f*)(b0 + k0);
        v2f fb1 = *(const v2f*)(b1 + k0);
        // next-block prefetch -> global_prefetch_b8 (speculative, OOB-safe)
        __builtin_prefetch(a0 + k0 + 64, 0, 3);
        __builtin_prefetch(b0 + k0 + 64, 0, 3);
        acc00 = __builtin_amdgcn_wmma_f32_16x16x4_f32(false, fa0, false, fb0,
                                                      (short)0, acc00, false, false);
        acc01 = __builtin_amdgcn_wmma_f32_16x16x4_f32(false, fa0, false, fb1,
                                                      (short)0, acc01, false, false);
        acc10 = __builtin_amdgcn_wmma_f32_16x16x4_f32(false, fa1, false, fb0,
                                                      (short)0, acc10, false, false);
        acc11 = __builtin_amdgcn_wmma_f32_16x16x4_f32(false, fa1, false, fb1,
                                                      (short)0, acc11, false, false);
    }

    const size_t rbase = (size_t)(mt * 32 + 8 * hi);
    const size_t cbase = (size_t)(nt * 32 + l);
    float* c00 = C + rbase * ldc + cbase;
    float* c01 = c00 + 16;
    float* c10 = c00 + (size_t)16 * ldc;
    float* c11 = c10 + 16;
    if (accumulate) {
        #pragma unroll
        for (int r = 0; r < 8; ++r) {
            c00[(size_t)r * ldc] += acc00[r];
            c01[(size_t)r * ldc] += acc01[r];
            c10[(size_t)r * ldc] += acc10[r];
            c11[(size_t)r * ldc] += acc11[r];
        }
    } else {
        #pragma unroll
        for (int r = 0; r < 8; ++r) {
            c00[(size_t)r * ldc] = acc00[r];
            c01[(size_t)r * ldc] = acc01[r];
            c10[(size_t)r * ldc] = acc10[r];
            c11[(size_t)r * ldc] = acc11[r];
        }
    }
}

// ---------------- layernorm (one row of width D per block) ----------------
__global__ void ln_kernel(const float* __restrict__ x,
                          const float* __restrict__ w,
                          const float* __restrict__ b,
                          float* __restrict__ out, int D) {
    __shared__ float s1[256];
    __shared__ float s2[256];
    const int row = blockIdx.x;
    const float* xr = x + (size_t)row * D;
    float sum = 0.f, sq = 0.f;
    for (int j = threadIdx.x; j < D; j += 256) {
        float v = xr[j]; sum += v; sq += v * v;
    }
    s1[threadIdx.x] = sum; s2[threadIdx.x] = sq;
    __syncthreads();
    for (int off = 128; off > 0; off >>= 1) {
        if (threadIdx.x < off) {
            s1[threadIdx.x] += s1[threadIdx.x + off];
            s2[threadIdx.x] += s2[threadIdx.x + off];
        }
        __syncthreads();
    }
    const float mean = s1[0] / (float)D;
    const float var  = s2[0] / (float)D - mean * mean;
    const float rs   = rsqrtf(var + EPS);
    float* outr = out + (size_t)row * D;
    for (int j = threadIdx.x; j < D; j += 256)
        outr[j] = (xr[j] - mean) * rs * w[j] + b[j];
}

// ---------------- causal depthwise conv (DC=4) + SiLU ----------------
__global__ void conv_silu_kernel(const float* __restrict__ xz,
                                 const float* __restrict__ cw,
                                 const float* __restrict__ cb,
                                 float* __restrict__ xc) {
    const int idx = blockIdx.x * blockDim.x + threadIdx.x;
    if (idx >= Bsz * Lseq * DI) return;
    const int d  = idx % DI;
    const int t  = (idx / DI) % Lseq;
    const int bb = idx / (DI * Lseq);
    const float* base = xz + (size_t)bb * Lseq * (2 * DI);
    float acc = cb[d];
    #pragma unroll
    for (int j = 0; j < DC; ++j) {
        int tt = t - (DC - 1) + j;
        if (tt >= 0) acc += cw[d * DC + j] * base[(size_t)tt * (2 * DI) + d];
    }
    xc[idx] = acc / (1.f + __expf(-acc));   // SiLU
}

// ---------------- selective scan (fused softplus + SiLU gate) ----------------
// One thread per (b, d) channel; 16-state recurrence in registers.
// B/C vectors (shared by all d of a batch) staged into LDS per T-chunk via
// GLOBAL_LOAD_ASYNC_TO_LDS_B128 (ASYNCcnt) when available.
#define TCHUNK 64
__global__ void scan_kernel(const float* __restrict__ xc,
                            const float* __restrict__ dtraw,
                            const float* __restrict__ dbc,
                            const float* __restrict__ A_log,
                            const float* __restrict__ Dp,
                            const float* __restrict__ dt_bias,
                            const float* __restrict__ xz,
                            float* __restrict__ y) {
    __shared__ float sBC[TCHUNK][2 * DS];   // [t][0:16)=B, [t][16:32)=C
    const int d  = blockIdx.x * 256 + threadIdx.x;
    const int bb = blockIdx.y;

    float Ad[DS];
    #pragma unroll
    for (int n = 0; n < DS; ++n)
        Ad[n] = -__expf(A_log[(size_t)d * DS + n]);
    const float Dd   = Dp[d];
    const float bias = dt_bias[d];

    float h[DS];
    #pragma unroll
    for (int n = 0; n < DS; ++n) h[n] = 0.f;

    for (int t0 = 0; t0 < Lseq; t0 += TCHUNK) {
        __syncthreads();
#if USE_ASYNC_LDS
        // 64 rows * 32 floats = 512 x 16B chunks; 2 per thread
        #pragma unroll
        for (int it = 0; it < 2; ++it) {
            const int c  = threadIdx.x + it * 256;
            const int tt = c >> 3;
            const int q  = c & 7;
            const float* g = dbc + (size_t)(bb * Lseq + t0 + tt) * (DR + 2 * DS)
                                 + DR + q * 4;
            __builtin_amdgcn_global_load_async_to_lds_b128(
                (gbl_v4i_ptr)g, (lds_v4i_ptr)&sBC[tt][q * 4], 0, 0);
        }
#if __has_builtin(__builtin_amdgcn_s_wait_asynccnt)
        __builtin_amdgcn_s_wait_asynccnt(0);
#else
        asm volatile("s_wait_asynccnt 0x0" ::: "memory");
#endif
#else
        for (int j = threadIdx.x; j < TCHUNK * 2 * DS; j += 256) {
            const int tt = j >> 5;
            const int c  = j & 31;
            sBC[tt][c] = dbc[(size_t)(bb * Lseq + t0 + tt) * (DR + 2 * DS) + DR + c];
        }
#endif
        __syncthreads();
        for (int t = 0; t < TCHUNK; ++t) {
            const size_t m = (size_t)bb * Lseq + t0 + t;
            float dtt = dtraw[m * DI + d] + bias;
            dtt = (dtt > 20.f) ? dtt : log1pf(__expf(dtt));     // softplus
            const float u  = xc[m * DI + d];
            const float du = dtt * u;
            float yy = 0.f;
            #pragma unroll
            for (int n = 0; n < DS; ++n) {
                h[n] = __expf(dtt * Ad[n]) * h[n] + du * sBC[t][n];
                yy  += h[n] * sBC[t][DS + n];
            }
            const float z = xz[m * (2 * DI) + DI + d];
            y[m * DI + d] = (yy + u * Dd) * (z / (1.f + __expf(-z)));  // gate
        }
    }
}

// ---------------- host-side orchestration ----------------
extern "C" void kernel_launch(void* const* d_in, const int* in_sizes, int n_in,
                              void* d_out, int out_size, void* d_ws, size_t ws_size,
                              hipStream_t stream) {
    const float* input     = (const float*)d_in[0];
    const float* norm_w    = (const float*)d_in[1];
    const float* norm_b    = (const float*)d_in[2];
    const float* in_proj_w = (const float*)d_in[3];
    const float* conv_w    = (const float*)d_in[4];
    const float* conv_b    = (const float*)d_in[5];
    const float* x_proj_w  = (const float*)d_in[6];
    const float* dt_proj_w = (const float*)d_in[7];
    const float* dt_proj_b = (const float*)d_in[8];
    const float* A_log     = (const float*)d_in[9];
    const float* Dvec      = (const float*)d_in[10];
    const float* out_proj_w= (const float*)d_in[11];
    const float* normf_w   = (const float*)d_in[12];
    const float* normf_b   = (const float*)d_in[13];

    float* ws = (float*)d_ws;
    size_t off = 0;
    float* R   = ws + off; off += (size_t)Mrows * DMd;            // residual
    float* H   = ws + off; off += (size_t)Mrows * DMd;            // post-LN hidden
    float* XZ  = ws + off; off += (size_t)Mrows * 2 * DI;         // in_proj out
    float* XC  = ws + off; off += (size_t)Mrows * DI;             // conv+silu out
    float* DBC = ws + off; off += (size_t)Mrows * (DR + 2 * DS);  // x_proj out
    float* DT  = ws + off; off += (size_t)Mrows * DI;             // raw dt
    float* Y   = ws + off; off += (size_t)Mrows * DI;             // gated scan out

    // residual = input
    (void)hipMemcpyAsync(R, input, (size_t)Mrows * DMd * sizeof(float),
                         hipMemcpyDeviceToDevice, stream);

    const dim3 gemmBlk(32, 4);
    const int elemBlk = 256;
    const int nElemDI = Mrows * DI;

    for (int i = 0; i < NL; ++i) {
        // H = LN(R)
        ln_kernel<<<Mrows, 256, 0, stream>>>(R, norm_w + (size_t)i * DMd,
                                             norm_b + (size_t)i * DMd, H, DMd);

        // XZ = H @ in_proj_w^T   (M=8192, N=4096, K=1024)
        {
            dim3 grid((2 * DI / 32 + 3) / 4, Mrows / 32);
            gemm_wmma_f32<<<grid, gemmBlk, 0, stream>>>(
                H, in_proj_w + (size_t)i * (2 * DI) * DMd, XZ,
                Mrows, 2 * DI, DMd, DMd, DMd, 2 * DI, 0);
        }

        // XC = silu(causal_conv(XZ[:, :DI]))
        conv_silu_kernel<<<(nElemDI + elemBlk - 1) / elemBlk, elemBlk, 0, stream>>>(
            XZ, conv_w + (size_t)i * DI * DC, conv_b + (size_t)i * DI, XC);

        // DBC = XC @ x_proj_w^T  (M=8192, N=96, K=2048)
        {
            dim3 grid(((DR + 2 * DS) / 32 + 3) / 4, Mrows / 32);
            gemm_wmma_f32<<<grid, gemmBlk, 0, stream>>>(
                XC, x_proj_w + (size_t)i * (DR + 2 * DS) * DI, DBC,
                Mrows, DR + 2 * DS, DI, DI, DI, DR + 2 * DS, 0);
        }

        // DT = DBC[:, :DR] @ dt_proj_w^T  (M=8192, N=2048, K=64; lda = 96)
        {
            dim3 grid((DI / 32 + 3) / 4, Mrows / 32);
            gemm_wmma_f32<<<grid, gemmBlk, 0, stream>>>(
                DBC, dt_proj_w + (size_t)i * DI * DR, DT,
                Mrows, DI, DR, DR + 2 * DS, DR, DI, 0);
        }

        // Y = silu_gate( selective_scan(XC, softplus(DT+b), B, C, A, D), z )
        {
            dim3 grid(DI / 256, Bsz);
            scan_kernel<<<grid, 256, 0, stream>>>(
                XC, DT, DBC, A_log + (size_t)i * DI * DS, Dvec + (size_t)i * DI,
                dt_proj_b + (size_t)i * DI, XZ, Y);
        }

        // R += Y @ out_proj_w^T  (M=8192, N=1024, K=2048, accumulate)
        {
            dim3 grid((DMd / 32 + 3) / 4, Mrows / 32);
            gemm_wmma_f32<<<grid, gemmBlk, 0, stream>>>(
                Y, out_proj_w + (size_t)i * DMd * DI, R,
                Mrows, DMd, DI, DI, DI, DMd, 1);
        }
    }

    // out = LN(R, normf)
    ln_kernel<<<Mrows, 256, 0, stream>>>(R, normf_w, normf_b, (float*)d_out, DMd);
}